// Attention_66288525246706
// MI455X (gfx1250) — compile-verified
//
#include <hip/hip_runtime.h>

typedef float v2f __attribute__((ext_vector_type(2)));
typedef float v8f __attribute__((ext_vector_type(8)));

#define B_ 8
#define T_ 128
#define D_ 256
#define LOG2E_ 1.4426950408889634f

// One block per (b,t). 256 threads = 8 wave32s.
// partial[b,t,q] = enc[b,t,q] * colsum[b,t,q] / rowsum[b,t,q]
// where M[p][q] = exp(dec[b,p]*enc[b,t,q]),
//       colsum[q] = sum_p M[p][q],  rowsum[q] = sum_r M[q][r].
// exp(d*e) = exp2((d*log2e)*e): one v_mul + one v_exp per element.
// Reduction adds ride the matrix pipe (V_WMMA_F32_16X16X4_F32 with ones).
// Two independent mul/exp/wmma chains are interleaved IN THE SAME LOOP BODY
// so the scheduler fills TRANS hazard slots with the sibling chain's work
// (and the LDS operand loads are shared between the chains).
__global__ __launch_bounds__(256) void attn_bt_kernel(const float* __restrict__ dec_t,
                                                      const float* __restrict__ enc_out,
                                                      float* __restrict__ partial)
{
    __shared__ float e_lds[D_];        // enc_out[b,t,:]          (raw)
    __shared__ float d_lds[D_];        // dec_t[b,:] * log2(e)    (pre-scaled)
    __shared__ float row_lds[D_];      // rowsum[q]
    __shared__ float colp_lds[8][D_];  // per-wave partial colsums

    const int bt   = blockIdx.x;       // b*T + t
    const int b    = bt >> 7;          // / T_
    const int tid  = threadIdx.x;
    const int lane = tid & 31;
    const int w    = tid >> 5;         // wave id 0..7
    const int m    = lane & 15;        // M/N index within half
    const int kh   = lane >> 4;        // which K-half this lane holds

    e_lds[tid] = enc_out[bt * D_ + tid];
    d_lds[tid] = dec_t[b * D_ + tid] * LOG2E_;
    __syncthreads();

    const v2f ones = {1.0f, 1.0f};

    // ---- row sums: wave w owns rows [32w, 32w+32) as two 16-row tiles,
    //      both chains interleaved per slab iteration ----
    // A-slab layout (16x4 f32): lane m = row, k = vgpr + 2*kh.
    // D = A x ones accumulates rowsum[p] replicated across the 16 N columns.
    {
        const float dp0 = d_lds[32 * w + m];
        const float dp1 = d_lds[32 * w + 16 + m];
        v8f acc0 = {}, acc1 = {};
        #pragma unroll 4
        for (int s = 0; s < 64; ++s) {          // 64 slabs of 4 q-values = 256
            const int q0 = 4 * s + 2 * kh;
            const float e0 = e_lds[q0];
            const float e1 = e_lds[q0 + 1];
            v2f a0, a1;
            a0.x = __builtin_amdgcn_exp2f(dp0 * e0);
            a0.y = __builtin_amdgcn_exp2f(dp0 * e1);
            a1.x = __builtin_amdgcn_exp2f(dp1 * e0);
            a1.y = __builtin_amdgcn_exp2f(dp1 * e1);
            acc0 = __builtin_amdgcn_wmma_f32_16x16x4_f32(false, a0, false, ones,
                                                         (short)0, acc0, false, false);
            acc1 = __builtin_amdgcn_wmma_f32_16x16x4_f32(false, a1, false, ones,
                                                         (short)0, acc1, false, false);
        }
        // C/D layout: lane L, VGPR r -> row = r + 8*(L>>4), replicated over n.
        if (m == 0) {                           // lanes 0 and 16 write 8 rows each
            const int rb0 = 32 * w + 8 * kh;
            #pragma unroll
            for (int r = 0; r < 8; ++r) row_lds[rb0 + r] = acc0[r];
            #pragma unroll
            for (int r = 0; r < 8; ++r) row_lds[rb0 + 16 + r] = acc1[r];
        }
    }

    // ---- partial column sums over this wave's 32 rows, for all 256 q,
    //      two q-tiles per iteration (independent chains, shared loads) ----
    // B-slab layout (4x16 f32): lane m = col, k = vgpr + 2*kh.
    // D = ones x B accumulates colsum partial replicated across the 16 M rows.
    for (int j = 0; j < 8; ++j) {
        const int q_base = 32 * j;
        const float eq0 = e_lds[q_base + m];
        const float eq1 = e_lds[q_base + 16 + m];
        v8f acc0 = {}, acc1 = {};
        #pragma unroll
        for (int s = 0; s < 8; ++s) {           // 8 slabs of 4 p-values = 32 rows
            const int p0 = 32 * w + 4 * s + 2 * kh;
            const float d0 = d_lds[p0];
            const float d1 = d_lds[p0 + 1];
            v2f b0, b1;
            b0.x = __builtin_amdgcn_exp2f(d0 * eq0);
            b0.y = __builtin_amdgcn_exp2f(d1 * eq0);
            b1.x = __builtin_amdgcn_exp2f(d0 * eq1);
            b1.y = __builtin_amdgcn_exp2f(d1 * eq1);
            acc0 = __builtin_amdgcn_wmma_f32_16x16x4_f32(false, ones, false, b0,
                                                         (short)0, acc0, false, false);
            acc1 = __builtin_amdgcn_wmma_f32_16x16x4_f32(false, ones, false, b1,
                                                         (short)0, acc1, false, false);
        }
        if (lane < 16) {                        // D[0][n], n = lane
            colp_lds[w][q_base + lane]      = acc0[0];
            colp_lds[w][q_base + 16 + lane] = acc1[0];
        }
    }

    __syncthreads();

    // ---- combine (fixed summation order -> deterministic) ----
    {
        const int q = tid;
        float cs = 0.f;
        #pragma unroll
        for (int ww = 0; ww < 8; ++ww) cs += colp_lds[ww][q];
        partial[bt * D_ + q] = e_lds[q] * cs / row_lds[q];
    }
}

// out[b,q] = sum_t partial[b,t,q]   (fixed order, coalesced across q)
__global__ __launch_bounds__(256) void attn_reduce_t(const float* __restrict__ partial,
                                                     float* __restrict__ out)
{
    const int b = blockIdx.x;
    const int q = threadIdx.x;
    float s = 0.f;
    #pragma unroll 8
    for (int t = 0; t < T_; ++t) s += partial[(b * T_ + t) * D_ + q];
    out[b * D_ + q] = s;
}

extern "C" void kernel_launch(void* const* d_in, const int* in_sizes, int n_in,
                              void* d_out, int out_size, void* d_ws, size_t ws_size,
                              hipStream_t stream)
{
    const float* dec_t   = (const float*)d_in[0];   // [B, D] f32
    const float* enc_out = (const float*)d_in[1];   // [B, T, D] f32
    float* out     = (float*)d_out;                 // [B, D] f32
    float* partial = (float*)d_ws;                  // B*T*D floats = 1 MiB scratch

    attn_bt_kernel<<<B_ * T_, 256, 0, stream>>>(dec_t, enc_out, partial);
    attn_reduce_t<<<B_, 256, 0, stream>>>(partial, out);
}